// AngularPenaltySMLoss_69269232549929
// MI455X (gfx1250) — compile-verified
//
#include <hip/hip_runtime.h>
#include <cstdint>

// ---------------------------------------------------------------------------
// ArcFace loss, MI455X (gfx1250, wave32).
//   Pass 1: one 256-thread block per row streams 50257 f32, sums exp(64*x),
//           computes the per-row ArcFace loss term into d_ws.
//   Pass 2: one block reduces 4096 row losses -> -mean into d_out[0].
// HBM-bound: ~823 MB @ 23.3 TB/s => ~35 us floor. exp via v_exp_f32.
// Wave reduction uses V_WMMA_F32_16X16X4_F32 (A = lane partials, B = ones).
// ---------------------------------------------------------------------------

typedef __attribute__((ext_vector_type(2))) float v2f;
typedef __attribute__((ext_vector_type(8))) float v8f;

#define KLOG2E     1.44269504088896340736f   // log2(e)
#define KS         64.0f                     // arcface s
#define KSEXP2     92.33248261689365807f     // 64 * log2(e)
#define KCOSM      0.87758256189037271612f   // cos(0.5)
#define KSINM      0.47942553860420300027f   // sin(0.5)
#define KLN2       0.69314718055994530942f   // ln(2)
#define KEPS       1e-7f
#define NT         256                       // threads per block (8 wave32)

// Sum a value across all 32 lanes of a wave using the matrix pipe.
// A(16x4) holds the 32 lane values (K=0 / K=2 slots, K=1/3 zeroed),
// B(4x16) = ones  =>  D[m,n] = lane_m + lane_{m+16}.  Each lane sums its 8
// D registers (half the rows of one column), then SWAPX16 adds the other half.
// EXEC must be all ones at the call site (uniform control flow).
__device__ __forceinline__ float wave_sum32(float v) {
    v2f a; a[0] = v;    a[1] = 0.0f;
    v2f b; b[0] = 1.0f; b[1] = 1.0f;
    v8f c = {};
    v8f d = __builtin_amdgcn_wmma_f32_16x16x4_f32(
        /*neg_a=*/false, a, /*neg_b=*/false, b,
        /*c_mod=*/(short)0, c, /*reuse_a=*/false, /*reuse_b=*/false);
    float p = ((d[0] + d[1]) + (d[2] + d[3])) + ((d[4] + d[5]) + (d[6] + d[7]));
    // group-of-32 swizzle: and=0x1f, or=0, xor=0x10  -> SWAPX16
    int q = __builtin_amdgcn_ds_swizzle(__float_as_int(p), 0x401f);
    return p + __int_as_float(q);
}

__global__ void __launch_bounds__(NT)
arcface_row_kernel(const float* __restrict__ wf,
                   const int*   __restrict__ labels,
                   float*       __restrict__ row_loss,
                   int C) {
    const int    row  = blockIdx.x;
    const int    tid  = threadIdx.x;
    const float* rp   = wf + (size_t)row * (size_t)C;

    // Peel to 16B alignment (row stride is odd -> odd rows are 4B aligned).
    int mis  = (int)(((uintptr_t)rp >> 2) & 3u);
    int head = (4 - mis) & 3;
    if (head > C) head = C;

    float acc = 0.0f;
    if (tid < head)
        acc += __builtin_amdgcn_exp2f(KSEXP2 * rp[tid]);

    const float4* bp  = (const float4*)(rp + head);
    const int     n4  = (C - head) >> 2;
    const int     rem = (C - head) & 3;

    for (int i = tid; i < n4; i += NT) {
        float4 x = bp[i];
        __builtin_prefetch(bp + i + 4 * NT, 0, 0);   // global_prefetch_b8, ~4 iters ahead
        acc += __builtin_amdgcn_exp2f(KSEXP2 * x.x);
        acc += __builtin_amdgcn_exp2f(KSEXP2 * x.y);
        acc += __builtin_amdgcn_exp2f(KSEXP2 * x.z);
        acc += __builtin_amdgcn_exp2f(KSEXP2 * x.w);
    }
    if (tid < rem)
        acc += __builtin_amdgcn_exp2f(KSEXP2 * rp[head + 4 * n4 + tid]);

    // Wave reduce (WMMA), then combine the 8 waves through LDS.
    float wsum = wave_sum32(acc);

    __shared__ float sred[NT / 32];
    const int wave = tid >> 5;
    if ((tid & 31) == 0) sred[wave] = wsum;
    __syncthreads();

    if (tid == 0) {
        float total = 0.0f;
        #pragma unroll
        for (int w = 0; w < NT / 32; ++w) total += sred[w];

        const float t  = rp[labels[row]];
        float tc = fminf(fmaxf(t, -1.0f + KEPS), 1.0f - KEPS);
        // cos(acos(tc) + m) = tc*cos(m) - sqrt(1-tc^2)*sin(m)
        float num = KS * (tc * KCOSM -
                          __builtin_sqrtf(fmaxf(1.0f - tc * tc, 0.0f)) * KSINM);
        float excl  = total - __builtin_amdgcn_exp2f(KSEXP2 * t);   // unclipped, as reference
        float denom = __builtin_amdgcn_exp2f(num * KLOG2E) + excl;
        float L     = num - __builtin_amdgcn_logf(denom) * KLN2;    // ln = log2 * ln2
        row_loss[row] = L;
    }
}

__global__ void __launch_bounds__(NT)
arcface_final_kernel(const float* __restrict__ row_loss,
                     float*       __restrict__ out,
                     int B) {
    const int tid = threadIdx.x;
    float acc = 0.0f;
    for (int i = tid; i < B; i += NT) acc += row_loss[i];

    float wsum = wave_sum32(acc);

    __shared__ float sred[NT / 32];
    if ((tid & 31) == 0) sred[tid >> 5] = wsum;
    __syncthreads();

    if (tid == 0) {
        float total = 0.0f;
        #pragma unroll
        for (int w = 0; w < NT / 32; ++w) total += sred[w];
        out[0] = -total / (float)B;
    }
}

extern "C" void kernel_launch(void* const* d_in, const int* in_sizes, int n_in,
                              void* d_out, int out_size, void* d_ws, size_t ws_size,
                              hipStream_t stream) {
    const float* wf     = (const float*)d_in[0];
    const int*   labels = (const int*)d_in[1];   // jax default x64-off => int32
    const int B = in_sizes[1];
    const int C = in_sizes[0] / B;

    float* row_loss = (float*)d_ws;              // B floats of scratch

    arcface_row_kernel<<<B, NT, 0, stream>>>(wf, labels, row_loss, C);
    arcface_final_kernel<<<1, NT, 0, stream>>>(row_loss, (float*)d_out, B);
}